// QuantumClassifier_38448547233944
// MI455X (gfx1250) — compile-verified
//
#include <hip/hip_runtime.h>
#include <hip/hip_bf16.h>
#include <math.h>

typedef __attribute__((ext_vector_type(2))) float v2f;
typedef __attribute__((ext_vector_type(8))) float v8f;

// ---------------------------------------------------------------------------
// Complex 2x2 gate applied to an amplitude pair (a, b): [a;b] = m * [a;b]
// ---------------------------------------------------------------------------
__device__ __forceinline__ void gate2(float& ar, float& ai, float& br, float& bi,
                                      float m00r, float m00i, float m01r, float m01i,
                                      float m10r, float m10i, float m11r, float m11i) {
    float nar = m00r * ar - m00i * ai + m01r * br - m01i * bi;
    float nai = m00r * ai + m00i * ar + m01r * bi + m01i * br;
    float nbr = m10r * ar - m10i * ai + m11r * br - m11i * bi;
    float nbi = m10r * ai + m10i * ar + m11r * bi + m11i * br;
    ar = nar; ai = nai; br = nbr; bi = nbi;
}

// ---------------------------------------------------------------------------
// Kernel 1: build the fixed 16x16 complex unitary M of the 8 shared blocks.
// Thread t (t<16) pushes basis state |t> through the 64 gates -> column t.
// ws[0..255]   = Re(M)[j*16+i]   (out[j] = sum_i M[j][i] * psi0[i])
// ws[256..511] = Im(M)[j*16+i]
// ---------------------------------------------------------------------------
__global__ void build_unitary_kernel(const float* __restrict__ u3p,
                                     const float* __restrict__ cu3p,
                                     float* __restrict__ ws) {
    int t = threadIdx.x;
    if (t >= 16) return;

    float pr[16], pi[16];
#pragma unroll
    for (int s = 0; s < 16; ++s) { pr[s] = (s == t) ? 1.f : 0.f; pi[s] = 0.f; }

    for (int blk = 0; blk < 8; ++blk) {
        // ---- 4 single-qubit U3 gates (qubit w -> bit 3-w of the flat index)
#pragma unroll
        for (int w = 0; w < 4; ++w) {
            const float th = u3p[(blk * 4 + w) * 3 + 0];
            const float ph = u3p[(blk * 4 + w) * 3 + 1];
            const float la = u3p[(blk * 4 + w) * 3 + 2];
            float ct = cosf(th * 0.5f), st = sinf(th * 0.5f);
            float cl = cosf(la), sl = sinf(la);
            float cp = cosf(ph), sp = sinf(ph);
            float cpl = cosf(ph + la), spl = sinf(ph + la);
            float m00r = ct,        m00i = 0.f;
            float m01r = -cl * st,  m01i = -sl * st;
            float m10r = cp * st,   m10i = sp * st;
            float m11r = cpl * ct,  m11i = spl * ct;
            const int b = 3 - w;
#pragma unroll
            for (int s = 0; s < 16; ++s) {
                if ((s >> b) & 1) continue;
                int s1 = s | (1 << b);
                gate2(pr[s], pi[s], pr[s1], pi[s1],
                      m00r, m00i, m01r, m01i, m10r, m10i, m11r, m11i);
            }
        }
        // ---- 4 CU3 gates: control i, target (i+1)%4
#pragma unroll
        for (int i = 0; i < 4; ++i) {
            const int tgt = (i + 1) & 3;
            const float th = cu3p[(blk * 4 + i) * 3 + 0];
            const float ph = cu3p[(blk * 4 + i) * 3 + 1];
            const float la = cu3p[(blk * 4 + i) * 3 + 2];
            float ct = cosf(th * 0.5f), st = sinf(th * 0.5f);
            float cl = cosf(la), sl = sinf(la);
            float cp = cosf(ph), sp = sinf(ph);
            float cpl = cosf(ph + la), spl = sinf(ph + la);
            float m00r = ct,        m00i = 0.f;
            float m01r = -cl * st,  m01i = -sl * st;
            float m10r = cp * st,   m10i = sp * st;
            float m11r = cpl * ct,  m11i = spl * ct;
            const int bc = 3 - i;
            const int bt = 3 - tgt;
#pragma unroll
            for (int s = 0; s < 16; ++s) {
                if (!((s >> bc) & 1)) continue;   // control must be 1
                if ((s >> bt) & 1) continue;      // target low index
                int s1 = s | (1 << bt);
                gate2(pr[s], pi[s], pr[s1], pi[s1],
                      m00r, m00i, m01r, m01i, m10r, m10i, m11r, m11i);
            }
        }
    }

#pragma unroll
    for (int j = 0; j < 16; ++j) {
        ws[j * 16 + t]       = pr[j];
        ws[256 + j * 16 + t] = pi[j];
    }
}

// ---------------------------------------------------------------------------
// Kernel 2: per-sample features + product state psi0, then psi = M * psi0
// via V_WMMA_F32_16X16X4_F32 (16 samples per wave tile), then expvals.
// Block: 128 threads (4 waves), 64 samples.
// ---------------------------------------------------------------------------
#define SAMP_PER_BLOCK 64
#define THREADS_MAIN   128

__global__ __launch_bounds__(THREADS_MAIN)
void qclass_main_kernel(const float* __restrict__ x,
                        const float* __restrict__ ws,
                        float* __restrict__ out) {
    __shared__ float sPsi[SAMP_PER_BLOCK][16][2];  // psi0 (re,im)  8 KB
    __shared__ float sProb[SAMP_PER_BLOCK][16];    // |psi|^2       4 KB

    const int t = threadIdx.x;
    const int blockBase = blockIdx.x * SAMP_PER_BLOCK;

    // ---------------- Phase A: features -> per-qubit vectors -> psi0 -------
    if (t < SAMP_PER_BLOCK) {
        const int b = blockBase + t;
        const float* img = x + (size_t)b * 10000;   // 100x100 image

        float feats[16];
#pragma unroll
        for (int k = 0; k < 16; ++k) feats[k] = 0.f;
#pragma unroll
        for (int r = 0; r < 3; ++r) {
#pragma unroll
            for (int q = 0; q < 12; ++q) {          // cols 0..47, float4 chunks
                float4 v = *(const float4*)(img + r * 100 + q * 4);
                feats[(q * 4 + 0) / 3] += v.x;
                feats[(q * 4 + 1) / 3] += v.y;
                feats[(q * 4 + 2) / 3] += v.z;
                feats[(q * 4 + 3) / 3] += v.w;
            }
        }
#pragma unroll
        for (int k = 0; k < 16; ++k) feats[k] *= (1.f / 9.f);

        // v_w = RY(f[12+w]) RX(f[8+w]) RZ(f[4+w]) RY(f[w]) |0>
        float vr[4][2], vi[4][2];
#pragma unroll
        for (int w = 0; w < 4; ++w) {
            float a1 = feats[w] * 0.5f,      a2 = feats[4 + w] * 0.5f;
            float a3 = feats[8 + w] * 0.5f,  a4 = feats[12 + w] * 0.5f;
            float c1 = cosf(a1), s1 = sinf(a1);
            float c2 = cosf(a2), s2 = sinf(a2);
            float c3 = cosf(a3), s3 = sinf(a3);
            float c4 = cosf(a4), s4 = sinf(a4);
            // after RZ
            float x0r = c1 * c2, x0i = -c1 * s2;
            float x1r = s1 * c2, x1i =  s1 * s2;
            // RX
            float u0r = c3 * x0r + s3 * x1i;
            float u0i = c3 * x0i - s3 * x1r;
            float u1r = s3 * x0i + c3 * x1r;
            float u1i = -s3 * x0r + c3 * x1i;
            // RY
            vr[w][0] = c4 * u0r - s4 * u1r;  vi[w][0] = c4 * u0i - s4 * u1i;
            vr[w][1] = s4 * u0r + c4 * u1r;  vi[w][1] = s4 * u0i + c4 * u1i;
        }
        // psi0[s] = v0[s>>3] * v1[(s>>2)&1] * v2[(s>>1)&1] * v3[s&1]
        float t01r[4], t01i[4], t23r[4], t23i[4];
#pragma unroll
        for (int a = 0; a < 2; ++a)
#pragma unroll
            for (int c = 0; c < 2; ++c) {
                t01r[a * 2 + c] = vr[0][a] * vr[1][c] - vi[0][a] * vi[1][c];
                t01i[a * 2 + c] = vr[0][a] * vi[1][c] + vi[0][a] * vr[1][c];
                t23r[a * 2 + c] = vr[2][a] * vr[3][c] - vi[2][a] * vi[3][c];
                t23i[a * 2 + c] = vr[2][a] * vi[3][c] + vi[2][a] * vr[3][c];
            }
#pragma unroll
        for (int s = 0; s < 16; ++s) {
            int hi = s >> 2, lo = s & 3;
            sPsi[t][s][0] = t01r[hi] * t23r[lo] - t01i[hi] * t23i[lo];
            sPsi[t][s][1] = t01r[hi] * t23i[lo] + t01i[hi] * t23r[lo];
        }
    }
    __syncthreads();

    // ---------------- Phase B: psi = M * psi0 via WMMA f32 16x16x4 ---------
    const int lane = t & 31;
    const int wave = t >> 5;
    const int tileBase = wave * 16;          // this wave's 16 samples
    const int n = lane & 15;
    const bool hiHalf = lane >= 16;

    // B fragments: B[k][n] = M[n][kglobal]; lanes 0-15 hold K=0/1, 16-31 K=2/3
    const float* Mr = ws;
    const float* Mi = ws + 256;
    v2f Br[4], Bi[4];
#pragma unroll
    for (int kk = 0; kk < 4; ++kk) {
        int i0 = kk * 4 + (hiHalf ? 2 : 0);
        Br[kk][0] = Mr[n * 16 + i0];
        Br[kk][1] = Mr[n * 16 + i0 + 1];
        Bi[kk][0] = Mi[n * 16 + i0];
        Bi[kk][1] = Mi[n * 16 + i0 + 1];
    }

    v8f Dr = {};  // Re(psi): D[m=sample][n=amp]
    v8f Di = {};  // Im(psi)
#pragma unroll
    for (int kk = 0; kk < 4; ++kk) {
        int i0 = kk * 4 + (hiHalf ? 2 : 0);
        int smp = tileBase + n;              // A row = sample (lane % 16)
        v2f Ar, Ai, Ani;
        Ar[0] = sPsi[smp][i0][0];     Ar[1] = sPsi[smp][i0 + 1][0];
        Ai[0] = sPsi[smp][i0][1];     Ai[1] = sPsi[smp][i0 + 1][1];
        Ani[0] = -Ai[0];              Ani[1] = -Ai[1];
        // complex: Dr += Ar*Br - Ai*Bi ; Di += Ar*Bi + Ai*Br
        Dr = __builtin_amdgcn_wmma_f32_16x16x4_f32(false, Ar,  false, Br[kk], (short)0, Dr, false, false);
        Dr = __builtin_amdgcn_wmma_f32_16x16x4_f32(false, Ani, false, Bi[kk], (short)0, Dr, false, false);
        Di = __builtin_amdgcn_wmma_f32_16x16x4_f32(false, Ar,  false, Bi[kk], (short)0, Di, false, false);
        Di = __builtin_amdgcn_wmma_f32_16x16x4_f32(false, Ai,  false, Br[kk], (short)0, Di, false, false);
    }

    // probs into LDS: lane holds amp j=n for samples v (+8 for hi half)
#pragma unroll
    for (int v = 0; v < 8; ++v) {
        int smp = tileBase + v + (hiHalf ? 8 : 0);
        sProb[smp][n] = Dr[v] * Dr[v] + Di[v] * Di[v];
    }
    __syncthreads();

    // ---------------- expvals: <Z_w> = sum_j (1-2*bit(j,3-w)) * p[j] -------
#pragma unroll
    for (int rep = 0; rep < 2; ++rep) {
        int task = rep * THREADS_MAIN + t;       // 64 samples * 4 wires = 256
        int smp = task >> 2;
        int w = task & 3;
        float e = 0.f;
#pragma unroll
        for (int j = 0; j < 16; ++j) {
            float sgn = ((j >> (3 - w)) & 1) ? -1.f : 1.f;
            e += sgn * sProb[smp][j];
        }
        out[(size_t)(blockBase + smp) * 4 + w] = e;
    }
}

// ---------------------------------------------------------------------------
extern "C" void kernel_launch(void* const* d_in, const int* in_sizes, int n_in,
                              void* d_out, int out_size, void* d_ws, size_t ws_size,
                              hipStream_t stream) {
    const float* x    = (const float*)d_in[0];   // (B,1,100,100) f32
    const float* u3p  = (const float*)d_in[1];   // (8,4,3) f32
    const float* cu3p = (const float*)d_in[2];   // (8,4,3) f32
    float* out = (float*)d_out;                  // (B,4) f32
    float* ws  = (float*)d_ws;                   // 512 floats used for M

    build_unitary_kernel<<<1, 32, 0, stream>>>(u3p, cu3p, ws);

    const int B = in_sizes[0] / 10000;           // 16384
    const int nBlocks = B / SAMP_PER_BLOCK;      // 256
    qclass_main_kernel<<<nBlocks, THREADS_MAIN, 0, stream>>>(x, ws, out);
}